// ParameterizedQuantumCircuit_64132451664144
// MI455X (gfx1250) — compile-verified
//
#include <hip/hip_runtime.h>

#define NQ 14
#define DIM 16384
#define NLAYERS 5
#define BATCH 2048
#define NTHREADS 256

typedef __attribute__((ext_vector_type(2))) float v2f;
typedef __attribute__((ext_vector_type(8))) float v8f;

struct cpx { float re, im; };
__device__ __forceinline__ cpx cmul(cpx a, cpx b) {
    return { a.re * b.re - a.im * b.im, a.re * b.im + a.im * b.re };
}
__device__ __forceinline__ cpx cadd(cpx a, cpx b) { return { a.re + b.re, a.im + b.im }; }

// U = Rz(g) @ Ry(b) @ Rx(a)   (2x2 complex)
__device__ __forceinline__ void u_full(float a, float b, float g, cpx u[2][2]) {
    float sa, ca, sb, cb, sg, cg;
    sincosf(0.5f * a, &sa, &ca);
    sincosf(0.5f * b, &sb, &cb);
    sincosf(0.5f * g, &sg, &cg);
    cpx Rx[2][2] = { { {ca, 0.f}, {0.f, -sa} }, { {0.f, -sa}, {ca, 0.f} } };
    cpx Ry[2][2] = { { {cb, 0.f}, {-sb, 0.f} }, { {sb, 0.f}, {cb, 0.f} } };
    // T = Ry @ Rx
    cpx t00 = cadd(cmul(Ry[0][0], Rx[0][0]), cmul(Ry[0][1], Rx[1][0]));
    cpx t01 = cadd(cmul(Ry[0][0], Rx[0][1]), cmul(Ry[0][1], Rx[1][1]));
    cpx t10 = cadd(cmul(Ry[1][0], Rx[0][0]), cmul(Ry[1][1], Rx[1][0]));
    cpx t11 = cadd(cmul(Ry[1][0], Rx[0][1]), cmul(Ry[1][1], Rx[1][1]));
    // Rz is diagonal: diag(cg - i sg, cg + i sg)
    cpx z0 = { cg, -sg }, z1 = { cg, sg };
    u[0][0] = cmul(z0, t00); u[0][1] = cmul(z0, t01);
    u[1][0] = cmul(z1, t10); u[1][1] = cmul(z1, t11);
}

__device__ __forceinline__ void u_rx(float a, cpx u[2][2]) {
    float sa, ca;
    sincosf(0.5f * a, &sa, &ca);
    u[0][0] = { ca, 0.f }; u[0][1] = { 0.f, -sa };
    u[1][0] = { 0.f, -sa }; u[1][1] = { ca, 0.f };
}

__device__ __forceinline__ void u_ident(cpx u[2][2]) {
    u[0][0] = { 1.f, 0.f }; u[0][1] = { 0.f, 0.f };
    u[1][0] = { 0.f, 0.f }; u[1][1] = { 1.f, 0.f };
}

__device__ __forceinline__ v8f wmma4(v2f a, v2f b, v8f c) {
    // D = A(16x4,f32) x B(4x16,f32) + C(16x16,f32)
    return __builtin_amdgcn_wmma_f32_16x16x4_f32(false, a, false, b, (short)0, c, false, false);
}

// Apply fused 16x16 complex unitary to interleaved state S via a real 32-dim
// contraction:  Dr = A'r x B',  Di = A'i x B'  with B'[2k]=re_k, B'[2k+1]=im_k.
// A'r[M][2k,2k+1] = ( Ur[M][k], -Ui[M][k] ),  A'i[M][2k,2k+1] = ( Ui[M][k], Ur[M][k] )
// are pre-staged row-major (16x32) in LDS by the build stage.
// Group's 4 qubits occupy index bits [ib+3 : ib]; state idx = hi*(16<<ib) + r*(1<<ib) + lo.
__device__ __forceinline__ void apply_group(int ib, int lane, int wave, v2f* S,
                                            const float* Apr, const float* Api) {
    const int M    = lane & 15;
    const int half = lane >> 4;
    // A fragments, one aligned ds_load_b64 each, reused across all 64 tiles:
    // frag kk, vgpr j holds A'[M][kk*4 + 2*half + j]   (ISA 7.12.2 f32 A layout)
    v2f Ar[8], Ai[8];
#pragma unroll
    for (int kk = 0; kk < 8; ++kk) {
        int off = M * 32 + kk * 4 + 2 * half;
        Ar[kk] = *reinterpret_cast<const v2f*>(Apr + off);
        Ai[kk] = *reinterpret_cast<const v2f*>(Api + off);
    }
    const int lom = (1 << ib) - 1;
    for (int it = 0; it < 8; ++it) {                // uniform trip count: no EXEC masking
        int t    = it * 8 + wave;                   // this wave's tile
        int col  = t * 16 + M;                      // lane's column (N = lane%16)
        int hi   = col >> ib;
        int lo   = col & lom;
        int base = (hi << (ib + 4)) + lo;           // + row*(1<<ib) selects the row
        v8f Dr = { 0, 0, 0, 0, 0, 0, 0, 0 };
        v8f Di = { 0, 0, 0, 0, 0, 0, 0, 0 };
#pragma unroll
        for (int kk = 0; kk < 8; ++kk) {
            int rc = kk * 2 + half;                 // complex row this lane supplies
            v2f Bf = S[base + (rc << ib)];          // (re,im) = B fragment, used directly
            Dr = wmma4(Ar[kk], Bf, Dr);
            Di = wmma4(Ai[kk], Bf, Di);
        }
        // D layout: VGPR m holds (M = m + 8*half, N = lane%16)
#pragma unroll
        for (int m = 0; m < 8; ++m) {
            int Mw = m + half * 8;
            v2f dv = { Dr[m], Di[m] };
            S[base + (Mw << ib)] = dv;              // ds_store_b64
        }
    }
}

__global__ __launch_bounds__(NTHREADS)
void pqc_kernel(const float* __restrict__ x, const float* __restrict__ theta,
                const float* __restrict__ lam, const float* __restrict__ w,
                float* __restrict__ out) {
    __shared__ v2f   S[DIM];          // interleaved (re,im) state: 128 KB
    __shared__ float Apr[16 * 32];    // staged A'r (2 KB)
    __shared__ float Api[16 * 32];    // staged A'i (2 KB)
    __shared__ float thetaS[252];
    __shared__ float lamS[70];
    __shared__ float xs[16];
    __shared__ float red[NTHREADS];

    const int tid  = threadIdx.x;
    const int b    = blockIdx.x;
    const int lane = tid & 31;
    const int wave = tid >> 5;

    if (tid < 252) thetaS[tid] = theta[tid];
    if (tid < 70)  lamS[tid]   = lam[tid];
    if (tid < NQ)  xs[tid]     = x[b * NQ + tid];
    for (int i = tid; i < DIM; i += NTHREADS) { v2f z = { 0.f, 0.f }; S[i] = z; }
    __syncthreads();
    if (tid == 0) { v2f one = { 1.f, 0.f }; S[0] = one; }   // |0...0>
    __syncthreads();

    const int gstart[4] = { 0, 4, 8, 10 };   // group 3 = I(x)I(x)U12(x)U13 on qubits 10..13
    const int r = tid >> 4, c = tid & 15;    // this thread's fused-matrix entry U[r][c]

    for (int l = 0; l <= NLAYERS; ++l) {
        // ---- variational layer l: 4 fused group unitaries ----
        for (int g = 0; g < 4; ++g) {
            {
                cpx u[4][2][2];
#pragma unroll
                for (int j = 0; j < 4; ++j) {
                    int q = gstart[g] + j;
                    if (g == 3 && j < 2) u_ident(u[j]);
                    else {
                        const float* ang = &thetaS[l * 42 + q * 3];
                        u_full(ang[0], ang[1], ang[2], u[j]);
                    }
                }
                cpx p = { 1.f, 0.f };
#pragma unroll
                for (int j = 0; j < 4; ++j)
                    p = cmul(p, u[j][(r >> (3 - j)) & 1][(c >> (3 - j)) & 1]);
                int o = r * 32 + 2 * c;
                v2f pr = { p.re, -p.im };                       // A'r pair
                v2f pi = { p.im,  p.re };                       // A'i pair
                *reinterpret_cast<v2f*>(Apr + o) = pr;          // ds_store_b64
                *reinterpret_cast<v2f*>(Api + o) = pi;          // ds_store_b64
            }
            __syncthreads();
            apply_group(10 - gstart[g], lane, wave, S, Apr, Api);
            __syncthreads();
        }
        if (l < NLAYERS) {
            // ---- ring ZZ sign (qubit q <-> bit 13-q; pairs = adjacent bits + cyclic) ----
            for (int i = tid; i < DIM; i += NTHREADS) {
                unsigned v = (unsigned)i;
                unsigned e = __popc(v & (v >> 1)) + (v & (v >> 13) & 1u);
                if (e & 1u) { v2f s = S[i]; v2f ns = { -s.x, -s.y }; S[i] = ns; }
            }
            __syncthreads();
            // ---- encoding layer l: batch-dependent Rx gates, same grouping ----
            for (int g = 0; g < 4; ++g) {
                {
                    cpx u[4][2][2];
#pragma unroll
                    for (int j = 0; j < 4; ++j) {
                        int q = gstart[g] + j;
                        if (g == 3 && j < 2) u_ident(u[j]);
                        else u_rx(xs[q] * lamS[l * NQ + q], u[j]);
                    }
                    cpx p = { 1.f, 0.f };
#pragma unroll
                    for (int j = 0; j < 4; ++j)
                        p = cmul(p, u[j][(r >> (3 - j)) & 1][(c >> (3 - j)) & 1]);
                    int o = r * 32 + 2 * c;
                    v2f pr = { p.re, -p.im };
                    v2f pi = { p.im,  p.re };
                    *reinterpret_cast<v2f*>(Apr + o) = pr;
                    *reinterpret_cast<v2f*>(Api + o) = pi;
                }
                __syncthreads();
                apply_group(10 - gstart[g], lane, wave, S, Apr, Api);
                __syncthreads();
            }
        }
    }

    // ---- <Z...Z> expectation: sum (-1)^popcount(i) * |amp_i|^2 ----
    float acc = 0.f;
    for (int i = tid; i < DIM; i += NTHREADS) {
        v2f s = S[i];
        float pp = s.x * s.x + s.y * s.y;
        acc += (__popc((unsigned)i) & 1) ? -pp : pp;
    }
    red[tid] = acc;
    __syncthreads();
    for (int s = NTHREADS / 2; s > 0; s >>= 1) {
        if (tid < s) red[tid] += red[tid + s];
        __syncthreads();
    }
    if (tid == 0) {
        float val = red[0];
        float l0 = val * w[0], l1 = val * w[1];   // BETA == 1
        float m  = fmaxf(l0, l1);
        float e0 = expf(l0 - m), e1 = expf(l1 - m);
        float inv = 1.f / (e0 + e1);
        out[2 * b + 0] = e0 * inv;
        out[2 * b + 1] = e1 * inv;
    }
}

extern "C" void kernel_launch(void* const* d_in, const int* in_sizes, int n_in,
                              void* d_out, int out_size, void* d_ws, size_t ws_size,
                              hipStream_t stream) {
    (void)in_sizes; (void)n_in; (void)out_size; (void)d_ws; (void)ws_size;
    const float* x     = (const float*)d_in[0];
    const float* theta = (const float*)d_in[1];
    const float* lam   = (const float*)d_in[2];
    const float* w     = (const float*)d_in[3];
    pqc_kernel<<<dim3(BATCH), dim3(NTHREADS), 0, stream>>>(x, theta, lam, w, (float*)d_out);
}